// DecoderLayer_51874615001327
// MI455X (gfx1250) — compile-verified
//
#include <hip/hip_runtime.h>

typedef __attribute__((ext_vector_type(2))) float v2f;
typedef __attribute__((ext_vector_type(8))) float v8f;

#define NN 4096   // 64*64 elements per node
#define PP 16     // parents per node
#define ACTIVE_THRESHOLD 12

__device__ __forceinline__ float fast_tanh(float x) {
#if __has_builtin(__builtin_amdgcn_tanhf)
    return __builtin_amdgcn_tanhf(x);   // single V_TANH_F32 trans op on gfx1250
#else
    return tanhf(x);
#endif
}

// One block (256 threads = 8 waves) per node k.
// Each wave handles 512 contiguous output elements of node k.
// Fully unrolled: 16 chunks x 32 elements. Per chunk: two 16x16x4 WMMA chains
// (4 WMMAs each, one per parent quad), A = weights broadcast across rows,
// B = gathered data tile. No arrays anywhere -> no dynamic-index waterfalls;
// all loads/stores use immediate offsets from per-stream base addresses.
__global__ __launch_bounds__(256) void decoder_wmma_kernel(
    const float* __restrict__ prev,          // (M, 4096) prev_outputs
    const unsigned char* __restrict__ act,   // (M,) prev_is_active (bool, 1B)
    const int* __restrict__ pidx,            // (K, 16) parent_indices
    const float* __restrict__ w,             // (K, 16)
    const float* __restrict__ bias,          // (K,)
    float* __restrict__ out,                 // K*4096 outputs, then K active flags
    int K)
{
    const int k    = blockIdx.x;
    const int tid  = threadIdx.x;
    const int lane = tid & 31;
    const int wave = tid >> 5;       // 0..7
    const int col  = lane & 15;      // output column within a 16-wide tile
    const bool hi  = lane >= 16;
    const int kp   = k * PP;

    // Node-uniform parent indices & weights as INDIVIDUAL scalars (no arrays,
    // so nothing can be turned into a dynamic-index select waterfall).
    const int i0  = pidx[kp +  0], i1  = pidx[kp +  1], i2  = pidx[kp +  2], i3  = pidx[kp +  3];
    const int i4  = pidx[kp +  4], i5  = pidx[kp +  5], i6  = pidx[kp +  6], i7  = pidx[kp +  7];
    const int i8  = pidx[kp +  8], i9  = pidx[kp +  9], i10 = pidx[kp + 10], i11 = pidx[kp + 11];
    const int i12 = pidx[kp + 12], i13 = pidx[kp + 13], i14 = pidx[kp + 14], i15 = pidx[kp + 15];

    const float w0  = w[kp +  0], w1  = w[kp +  1], w2  = w[kp +  2], w3  = w[kp +  3];
    const float w4  = w[kp +  4], w5  = w[kp +  5], w6  = w[kp +  6], w7  = w[kp +  7];
    const float w8  = w[kp +  8], w9  = w[kp +  9], w10 = w[kp + 10], w11 = w[kp + 11];
    const float w12 = w[kp + 12], w13 = w[kp + 13], w14 = w[kp + 14], w15 = w[kp + 15];

    // Activity gate: >= 12 of 16 parents active.
    int na = 0;
    na += (act[i0]  != 0); na += (act[i1]  != 0); na += (act[i2]  != 0); na += (act[i3]  != 0);
    na += (act[i4]  != 0); na += (act[i5]  != 0); na += (act[i6]  != 0); na += (act[i7]  != 0);
    na += (act[i8]  != 0); na += (act[i9]  != 0); na += (act[i10] != 0); na += (act[i11] != 0);
    na += (act[i12] != 0); na += (act[i13] != 0); na += (act[i14] != 0); na += (act[i15] != 0);
    const float gate = (na >= ACTIVE_THRESHOLD) ? 1.0f : 0.0f;
    if (tid == 0) out[(size_t)K * NN + k] = gate;

    const float bk = bias[k];
    const int base = wave * 512;     // this wave's element range within node k

    // Per-lane A operands (weights) and gather-stream base addresses.
    // ISA layout (f32, 16x16x4): lanes 0-15 carry K={4g+0,4g+1} in VGPR{0,1};
    // lanes 16-31 carry K={4g+2,4g+3}. Each select is one v_cndmask; each
    // address is one shl/add chain off the selected index.
    v2f a0, a1, a2, a3;
    a0.x = hi ? w2  : w0;   a0.y = hi ? w3  : w1;
    a1.x = hi ? w6  : w4;   a1.y = hi ? w7  : w5;
    a2.x = hi ? w10 : w8;   a2.y = hi ? w11 : w9;
    a3.x = hi ? w14 : w12;  a3.y = hi ? w15 : w13;

    const int bc = base + col;
    const float* px0 = prev + (size_t)(hi ? i2  : i0 ) * NN + bc;
    const float* py0 = prev + (size_t)(hi ? i3  : i1 ) * NN + bc;
    const float* px1 = prev + (size_t)(hi ? i6  : i4 ) * NN + bc;
    const float* py1 = prev + (size_t)(hi ? i7  : i5 ) * NN + bc;
    const float* px2 = prev + (size_t)(hi ? i10 : i8 ) * NN + bc;
    const float* py2 = prev + (size_t)(hi ? i11 : i9 ) * NN + bc;
    const float* px3 = prev + (size_t)(hi ? i14 : i12) * NN + bc;
    const float* py3 = prev + (size_t)(hi ? i15 : i13) * NN + bc;

    float* po = out + (size_t)k * NN + base + lane;

    // One parent-quad step: 2 WMMAs (dual output tiles) on gathered B data.
    auto quad = [&](v2f aa, const float* px, const float* py, int o,
                    v8f& c0, v8f& c1) __attribute__((always_inline)) {
        v2f b0, b1;
        b0.x = px[o];
        b0.y = py[o];
        b1.x = px[o + 16];
        b1.y = py[o + 16];
        // D = A(16x4 weights, rows replicated) x B(4x16 data) + C
        c0 = __builtin_amdgcn_wmma_f32_16x16x4_f32(
                 false, aa, false, b0, (short)0, c0, false, false);
        c1 = __builtin_amdgcn_wmma_f32_16x16x4_f32(
                 false, aa, false, b1, (short)0, c1, false, false);
    };

#pragma unroll
    for (int it = 0; it < 16; ++it) {
        const int o = it * 32;           // constant after unroll -> imm offsets

        v8f c0 = { bk, bk, bk, bk, bk, bk, bk, bk };  // C preloaded with bias
        v8f c1 = c0;

        quad(a0, px0, py0, o, c0, c1);
        quad(a1, px1, py1, o, c0, c1);
        quad(a2, px2, py2, o, c0, c1);
        quad(a3, px3, py3, o, c0, c1);

        // All D rows are identical; VGPR0 of each accumulator holds
        // out[chunk_base + col]. Lanes 0-15 -> tile0, lanes 16-31 -> tile1,
        // i.e. exactly out[o + lane]: a fully coalesced 128B store.
        float v = hi ? c1[0] : c0[0];
        v = fast_tanh(v) * gate;         // multiply-gate keeps EXEC all-1s
        // Write-once output: non-temporal so it doesn't evict the gathered
        // prev_outputs working set from WGP$/L2.
        __builtin_nontemporal_store(v, po + o);
    }
}

extern "C" void kernel_launch(void* const* d_in, const int* in_sizes, int n_in,
                              void* d_out, int out_size, void* d_ws, size_t ws_size,
                              hipStream_t stream) {
    const float*         prev = (const float*)d_in[0];          // (M,64,64)
    const unsigned char* act  = (const unsigned char*)d_in[1];  // (M,) bool
    const int*           pidx = (const int*)d_in[2];            // (K,16)
    const float*         w    = (const float*)d_in[3];          // (K,16)
    const float*         b    = (const float*)d_in[4];          // (K,)
    float* out = (float*)d_out;

    const int K = in_sizes[4];   // bias length == number of nodes

    decoder_wmma_kernel<<<K, 256, 0, stream>>>(prev, act, pidx, w, b, out, K);
}